// ECC_60627758350829
// MI455X (gfx1250) — compile-verified
//
#include <hip/hip_runtime.h>
#include <stdint.h>

typedef __attribute__((ext_vector_type(2))) float v2f;
typedef __attribute__((ext_vector_type(8))) float v8f;

#define FI 128
#define FO 128
#define NTYPES 7

// ---------------------------------------------------------------------------
// GEMM: h[M x 128] = x[M x 128] @ W[128 x 128] + bias, one edge type at a time.
// Block = 256 threads = 8 wave32; each wave owns a 16-row M-tile and sweeps all
// 8 N-tiles with V_WMMA_F32_16X16X4_F32 (K stepped by 4, 32 steps).
//  - A tile (16x128 f32 = 64 VGPRs) preloaded with 32 pipelined global_load_b64
//  - W staged in LDS pre-paired as float2: Wp[k/2][n] = (W[k][n], W[k+1][n]),
//    so each B fragment is ONE aligned ds_load_b64 straight into the WMMA
//    operand pair (no repack movs), addressed by immediate offsets.
//  - B fragments double-buffered one k-step ahead so WMMAs never wait on
//    dscnt==0 (LDS latency hidden behind 8 in-flight matrix ops).
// ---------------------------------------------------------------------------
__global__ __launch_bounds__(256) void ecc_gemm(const float* __restrict__ x,
                                                const float* __restrict__ W,    // [K=128][N=128] row-major
                                                const float* __restrict__ bias, // [128]
                                                float* __restrict__ h,
                                                int M) {
    __shared__ v2f Wp[(FI / 2) * FO];        // 64 kp x 128 n x 8B = 64 KB

    const int tid = threadIdx.x;

    // Cooperative paired fill: group g -> kp = g/32, n4 = (g%32)*4.
    // Coalesced float4 reads of rows 2kp and 2kp+1, interleaved float2 writes.
#pragma unroll
    for (int i = 0; i < 8; ++i) {
        const int g  = tid + i * 256;        // 2048 groups total
        const int kp = g >> 5;
        const int n0 = (g & 31) * 4;
        const float4 r0 = *(const float4*)(W + (2 * kp    ) * FO + n0);
        const float4 r1 = *(const float4*)(W + (2 * kp + 1) * FO + n0);
        v2f* dst = &Wp[kp * FO + n0];
        dst[0] = (v2f){r0.x, r1.x};
        dst[1] = (v2f){r0.y, r1.y};
        dst[2] = (v2f){r0.z, r1.z};
        dst[3] = (v2f){r0.w, r1.w};
    }

    const int wave = tid >> 5;
    const int lane = tid & 31;
    const int lh   = lane >> 4;   // 0..1  (K-half / M-half selector)
    const int lm   = lane & 15;   // 0..15 (M for A, N for B/C)

    const int tileM0 = blockIdx.x * 128 + wave * 16;
    int row  = tileM0 + lm;
    int rowc = row < M ? row : (M - 1);     // clamp: clamped rows never stored
    const float* xrow = x + (size_t)rowc * FI;

    // ---- Preload entire A tile into registers (overlaps W->LDS fill) ------
    // areg[kk] = A[M=lm, K = 4*kk + 2*lh .. +1]
    v2f areg[32];
#pragma unroll
    for (int kk = 0; kk < 32; ++kk)
        areg[kk] = *(const v2f*)(xrow + 4 * kk + 2 * lh);

    __syncthreads();   // LDS W ready

    v8f acc[8];
#pragma unroll
    for (int nt = 0; nt < 8; ++nt) { v8f z = {}; acc[nt] = z; }

    // B fragment for step kk: kp = 2*kk + lh, element n = nt*16 + lm.
    const v2f* Wlane = &Wp[lh * FO + lm];    // + kk*(2*FO) + nt*16 as imm offs

    v2f bcur[8];
#pragma unroll
    for (int nt = 0; nt < 8; ++nt) bcur[nt] = Wlane[nt * 16];

#pragma unroll
    for (int kk = 0; kk < 32; ++kk) {
        v2f bn[8];
        if (kk + 1 < 32) {                   // prefetch next step's fragments
#pragma unroll
            for (int nt = 0; nt < 8; ++nt)
                bn[nt] = Wlane[(kk + 1) * (2 * FO / 2) * 2 + nt * 16];
        }
#pragma unroll
        for (int nt = 0; nt < 8; ++nt)
            acc[nt] = __builtin_amdgcn_wmma_f32_16x16x4_f32(
                false, areg[kk], false, bcur[nt], (short)0, acc[nt], false, false);
        if (kk + 1 < 32) {
#pragma unroll
            for (int nt = 0; nt < 8; ++nt) bcur[nt] = bn[nt];
        }
    }

    // Epilogue: D[M = r + 8*lh, N = nt*16+lm] + bias
    const int mBase = tileM0 + 8 * lh;
#pragma unroll
    for (int nt = 0; nt < 8; ++nt) {
        const float bv = bias[nt * 16 + lm];
#pragma unroll
        for (int r = 0; r < 8; ++r) {
            const int mo = mBase + r;
            if (mo < M)
                h[(size_t)mo * FO + nt * 16 + lm] = acc[nt][r] + bv;
        }
    }
}

// ---------------------------------------------------------------------------
// Edge kernel: one wave per (batch, edge). Coalesced 512B gather of h row,
// scale by edge weight, hardware fp32 atomic scatter-add (4 per lane).
// ---------------------------------------------------------------------------
__device__ __forceinline__ void atomic_add_f32_hw(float* p, float v) {
    // no-return global fp32 atomic add (tracked on STOREcnt; drained at endpgm)
    asm volatile("global_atomic_add_f32 %0, %1, off"
                 :: "v"((uint64_t)(uintptr_t)p), "v"(v)
                 : "memory");
}

__global__ __launch_bounds__(256) void ecc_edge(const float* __restrict__ h,
                                                const float* __restrict__ adjv,
                                                const int*   __restrict__ endn,
                                                const int*   __restrict__ startn,
                                                float* __restrict__ out,
                                                int E, int N) {
    const int w    = blockIdx.x * 8 + (threadIdx.x >> 5);  // wave id
    const int lane = threadIdx.x & 31;
    const int total = 2 * E;                               // B = 2
    if (w >= total) return;                                // wave-uniform exit

    const int batch = (w >= E) ? 1 : 0;
    const int e     = w - batch * E;

    const int   en = endn[e];
    const int   sn = startn[e];
    const float a  = adjv[e];

    // Gather: 32 lanes x float4 = one contiguous 512B row of h (L2-resident).
    const float4 hv =
        ((const float4*)(h + ((size_t)batch * N + (size_t)en) * FO))[lane];

    float* p = out + ((size_t)batch * N + (size_t)sn) * FO + lane * 4;
    atomic_add_f32_hw(p + 0, a * hv.x);
    atomic_add_f32_hw(p + 1, a * hv.y);
    atomic_add_f32_hw(p + 2, a * hv.z);
    atomic_add_f32_hw(p + 3, a * hv.w);
}

// ---------------------------------------------------------------------------
// inputs: [0]=x [B,N,128] f32   [1]=W [7,128,128] f32   [2]=b [7,128] f32
//         [3]=adj_values [7,E] f32  [4]=end_nodes [7,E] i32  [5]=start_nodes [7,E] i32
// out: [B,N,128] f32.  ws: one h buffer, B*N*128 floats (~102 MB).
// ---------------------------------------------------------------------------
extern "C" void kernel_launch(void* const* d_in, const int* in_sizes, int n_in,
                              void* d_out, int out_size, void* d_ws, size_t ws_size,
                              hipStream_t stream) {
    const float* x      = (const float*)d_in[0];
    const float* W      = (const float*)d_in[1];
    const float* b      = (const float*)d_in[2];
    const float* adjv   = (const float*)d_in[3];
    const int*   endn   = (const int*)d_in[4];
    const int*   startn = (const int*)d_in[5];
    float*       out    = (float*)d_out;
    float*       h      = (float*)d_ws;

    const int E = in_sizes[3] / NTYPES;
    const int M = in_sizes[0] / FI;   // B*N rows
    const int B = 2;                  // per reference
    const int N = M / B;

    hipMemsetAsync(out, 0, (size_t)out_size * sizeof(float), stream);

    const int gemmBlocks = (M + 127) / 128;
    const int edgeBlocks = (2 * E + 7) / 8;

    // One type at a time: keeps h (102MB) + out (102MB) ~L2-resident, and the
    // in-order stream serializes gemm->edge so the single h buffer is safe.
    for (int t = 0; t < NTYPES; ++t) {
        ecc_gemm<<<gemmBlocks, 256, 0, stream>>>(
            x, W + (size_t)t * FI * FO, b + (size_t)t * FO, h, M);
        ecc_edge<<<edgeBlocks, 256, 0, stream>>>(
            h, adjv + (size_t)t * E, endn + (size_t)t * E,
            startn + (size_t)t * E, out, E, N);
    }
}